// YoloV1Loss_8959301780125
// MI455X (gfx1250) — compile-verified
//
#include <hip/hip_runtime.h>
#include <hip/hip_bf16.h>
#include <stdint.h>

// ---------------------------------------------------------------------------
// YOLOv1 loss on MI455X (gfx1250).
// Memory-bound reduction: 192.7 MB in -> 4 floats out. Strategy:
//   * TDM (tensor_load_to_lds) streams 15 KB/tile/input Global->LDS, double
//     buffered, tracked with TENSORcnt (s_wait_tensorcnt).
//   * 128 threads/block, 1 cell/thread/tile from LDS (conflict-free stride-30).
//   * Deterministic two-pass reduction (no float atomics).
// ---------------------------------------------------------------------------

#define CELLF (1.0f / 7.0f)
#define LCF 5.0f
#define LNF 0.5f

constexpr int kCells = 16384 * 7 * 7;      // 802816 cells
constexpr int kTileC = 128;                 // cells per tile == blockDim.x
constexpr int kTiles = kCells / kTileC;     // 6272 (exact)
constexpr int kTileF = kTileC * 30;         // 3840 floats per input per tile
constexpr int kTileB = kTileF * 4;          // 15360 bytes
constexpr int kGrid  = 1568;                // 6272 / 1568 = 4 tiles per block

typedef unsigned int v4u __attribute__((ext_vector_type(4)));
typedef int          v8i __attribute__((ext_vector_type(8)));
typedef int          v4i __attribute__((ext_vector_type(4)));

// Build a D# for a contiguous 1-D dword copy (ndw dwords) Global -> LDS and
// issue TENSOR_LOAD_TO_LDS. Layout per CDNA5 ISA ch.8 (group0/group1).
__device__ __forceinline__ void tdm_load_1d(const float* gsrc,
                                            unsigned lds_byte, unsigned ndw) {
  uint64_t ga = (uint64_t)(uintptr_t)gsrc;
  v4u g0;
  g0[0] = 1u;                                              // count=1, user mode
  g0[1] = lds_byte;                                        // lds_addr (bytes)
  g0[2] = (unsigned)ga;                                    // global_addr[31:0]
  g0[3] = (unsigned)((ga >> 32) & 0x01FFFFFFu)             // global_addr[56:32]
          | (2u << 30);                                    // type=2 ("image")
  v8i g1;
  g1[0] = (int)(2u << 16);                                 // data_size=2 -> 4B
  g1[1] = (int)((ndw & 0xFFFFu) << 16);                    // tensor_dim0 lo16
  g1[2] = (int)((ndw >> 16) | (1u << 16));                 // td0 hi16 | td1=1
  g1[3] = (int)((ndw & 0xFFFFu) << 16);                    // tile_dim0
  g1[4] = 1;                                               // tile_dim1=1, tile_dim2=0
  g1[5] = (int)ndw;                                        // tensor_dim0_stride lo32
  g1[6] = 0;                                               // stride hi | td1_stride lo
  g1[7] = 0;
  v4i z4 = {0, 0, 0, 0};
#if __clang_major__ >= 23
  v8i z8 = {0, 0, 0, 0, 0, 0, 0, 0};
  __builtin_amdgcn_tensor_load_to_lds(g0, g1, z4, z4, z8, 0);
#else
  __builtin_amdgcn_tensor_load_to_lds(g0, g1, z4, z4, 0);
#endif
}

__global__ __launch_bounds__(kTileC) void yolo_partial(
    const float* __restrict__ pred, const float* __restrict__ targ,
    float* __restrict__ partial) {
  // [buf0: pred(3840) targ(3840)][buf1: pred targ][16-float reduction scratch]
  __shared__ float smem[4 * kTileF + 16];
  const int tid = threadIdx.x;
  const int wv  = tid >> 5;                 // wave id (wave32)
  const int G   = gridDim.x;

  float a_cls = 0.f, a_conf = 0.f, a_coord = 0.f, a_noobj = 0.f;

  int tile = blockIdx.x;
  // Prologue: prefetch first tile into buffer 0 (TDM ignores EXEC; the
  // wave-uniform branch ensures exactly one issue per block).
  if (wv == 0 && tile < kTiles) {
    tdm_load_1d(pred + (size_t)tile * kTileF, 0u, kTileF);
    tdm_load_1d(targ + (size_t)tile * kTileF, (unsigned)kTileB, kTileF);
  }

  int it = 0;
  for (; tile < kTiles; tile += G, ++it) {
    const int buf = it & 1;
    const int nxt = tile + G;

    __syncthreads();  // everyone done with buffer buf^1 -> safe to refill it
    if (nxt < kTiles && wv == 0) {
      const unsigned pb = (unsigned)((buf ^ 1) * 2 * kTileB);
      tdm_load_1d(pred + (size_t)nxt * kTileF, pb, kTileF);
      tdm_load_1d(targ + (size_t)nxt * kTileF, pb + (unsigned)kTileB, kTileF);
    }
    // Retire the oldest descriptor pair (in-order): current buffer resident.
    if (nxt < kTiles) __builtin_amdgcn_s_wait_tensorcnt(2);
    else              __builtin_amdgcn_s_wait_tensorcnt(0);
    __syncthreads();  // release all waves onto the current buffer

    const float* cp = &smem[buf * 2 * kTileF + tid * 30];
    const float* ct = &smem[buf * 2 * kTileF + kTileF + tid * 30];

    float p[10], tb[10];
#pragma unroll
    for (int j = 0; j < 10; ++j) p[j] = cp[j];
#pragma unroll
    for (int j = 0; j < 10; ++j) tb[j] = ct[j];

    const float t4    = tb[4];
    const float obj   = (t4 == 1.0f) ? 1.0f : 0.0f;
    const float noobj = (t4 == 0.0f) ? 1.0f : 0.0f;
    {
      const float d0 = p[4] - tb[4], d1 = p[9] - tb[9];
      a_noobj += noobj * (d0 * d0 + d1 * d1);
    }

    // Target corners (faithful to the reference's in-place x2y2 bug).
    const float tw  = tb[2] * tb[2], th = tb[3] * tb[3];
    const float tx1 = tb[0] * CELLF - 0.5f * tw;
    const float ty1 = tb[1] * CELLF - 0.5f * th;
    const float tx2 = tx1 * CELLF + 0.5f * tw;
    const float ty2 = ty1 * CELLF + 0.5f * th;
    const float at  = (tx2 - tx1) * (ty2 - ty1);

    float iou[2];
#pragma unroll
    for (int bx = 0; bx < 2; ++bx) {
      const int o   = bx * 5;
      const float w = p[o + 2] * p[o + 2], h = p[o + 3] * p[o + 3];
      const float x1 = p[o + 0] * CELLF - 0.5f * w;
      const float y1 = p[o + 1] * CELLF - 0.5f * h;
      const float x2 = x1 * CELLF + 0.5f * w;
      const float y2 = y1 * CELLF + 0.5f * h;
      const float lx = fmaxf(x1, tx1), ly = fmaxf(y1, ty1);
      const float rx = fminf(x2, tx2), ry = fminf(y2, ty2);
      const float inter = fmaxf(rx - lx, 0.f) * fmaxf(ry - ly, 0.f);
      const float ap = (x2 - x1) * (y2 - y1);
      iou[bx] = inter / (ap + at - inter);
    }
    // argmax with numpy NaN-is-max semantics, first-occurrence tie break.
    const bool n0 = (iou[0] != iou[0]), n1 = (iou[1] != iou[1]);
    const int idx = (!n0 && (n1 || iou[1] > iou[0])) ? 1 : 0;
    const int o = idx * 5;
    {
      const float dx = p[o + 0] - tb[o + 0], dy = p[o + 1] - tb[o + 1];
      const float dw = p[o + 2] - tb[o + 2], dh = p[o + 3] - tb[o + 3];
      const float dc = p[o + 4] - tb[o + 4];
      a_coord += obj * (dx * dx + dy * dy + dw * dw + dh * dh);
      a_conf  += obj * dc * dc;
    }
    // class column: first argmax of targets[10:30]
    int   k    = 10;
    float best = ct[10];
#pragma unroll
    for (int j = 11; j < 30; ++j) {
      const float v = ct[j];
      if (v > best) { best = v; k = j; }
    }
    const float dk = cp[k] - ct[k];
    a_cls += obj * dk * dk;
  }

  a_cls *= LCF; a_conf *= LCF; a_coord *= LCF; a_noobj *= LNF;

  // wave32 reduction
#pragma unroll
  for (int off = 16; off > 0; off >>= 1) {
    a_cls   += __shfl_down(a_cls, off);
    a_conf  += __shfl_down(a_conf, off);
    a_coord += __shfl_down(a_coord, off);
    a_noobj += __shfl_down(a_noobj, off);
  }
  float* red = &smem[4 * kTileF];
  __syncthreads();
  if ((tid & 31) == 0) {
    red[wv * 4 + 0] = a_cls;  red[wv * 4 + 1] = a_conf;
    red[wv * 4 + 2] = a_coord; red[wv * 4 + 3] = a_noobj;
  }
  __syncthreads();
  if (tid == 0) {
    float c0 = 0.f, c1 = 0.f, c2 = 0.f, c3 = 0.f;
#pragma unroll
    for (int w = 0; w < kTileC / 32; ++w) {
      c0 += red[4 * w + 0]; c1 += red[4 * w + 1];
      c2 += red[4 * w + 2]; c3 += red[4 * w + 3];
    }
    partial[blockIdx.x * 4 + 0] = c0;
    partial[blockIdx.x * 4 + 1] = c1;
    partial[blockIdx.x * 4 + 2] = c2;
    partial[blockIdx.x * 4 + 3] = c3;
  }
}

__global__ __launch_bounds__(256) void yolo_final(
    const float* __restrict__ partial, int nblk, float* __restrict__ out) {
  __shared__ float red[8 * 4];
  float c0 = 0.f, c1 = 0.f, c2 = 0.f, c3 = 0.f;
  const float4* p4 = (const float4*)partial;  // 16B-aligned d_ws
  for (int i = threadIdx.x; i < nblk; i += blockDim.x) {
    const float4 v = p4[i];
    c0 += v.x; c1 += v.y; c2 += v.z; c3 += v.w;
  }
#pragma unroll
  for (int off = 16; off > 0; off >>= 1) {
    c0 += __shfl_down(c0, off); c1 += __shfl_down(c1, off);
    c2 += __shfl_down(c2, off); c3 += __shfl_down(c3, off);
  }
  const int wv = threadIdx.x >> 5;
  if ((threadIdx.x & 31) == 0) {
    red[wv * 4 + 0] = c0; red[wv * 4 + 1] = c1;
    red[wv * 4 + 2] = c2; red[wv * 4 + 3] = c3;
  }
  __syncthreads();
  if (threadIdx.x == 0) {
    float s0 = 0.f, s1 = 0.f, s2 = 0.f, s3 = 0.f;
#pragma unroll
    for (int w = 0; w < 8; ++w) {
      s0 += red[4 * w + 0]; s1 += red[4 * w + 1];
      s2 += red[4 * w + 2]; s3 += red[4 * w + 3];
    }
    out[0] = s0;                 // cls_loss
    out[1] = s1;                 // conf_loss
    out[2] = s2;                 // center_loss + wh_loss
    out[3] = s0 + s1 + s2 + s3;  // total (incl. noobj)
  }
}

extern "C" void kernel_launch(void* const* d_in, const int* in_sizes, int n_in,
                              void* d_out, int out_size, void* d_ws,
                              size_t ws_size, hipStream_t stream) {
  (void)in_sizes; (void)n_in; (void)out_size; (void)ws_size;
  const float* pred = (const float*)d_in[0];
  const float* targ = (const float*)d_in[1];
  float* out     = (float*)d_out;
  float* partial = (float*)d_ws;  // kGrid*4 floats = 25,088 B of scratch
  yolo_partial<<<kGrid, kTileC, 0, stream>>>(pred, targ, partial);
  yolo_final<<<1, 256, 0, stream>>>(partial, kGrid, out);
}